// SimGraphConstruction_87548613362340
// MI455X (gfx1250) — compile-verified
//
#include <hip/hip_runtime.h>
#include <hip/hip_bf16.h>
#include <float.h>

typedef __attribute__((ext_vector_type(2))) float v2f;
typedef __attribute__((ext_vector_type(8))) float v8f;

#define N_NODES 16384
#define DIM 512
#define TOPK 16
#define ROWS_PER_BLOCK 16
#define PANEL 128
#define KCHUNK 32
#define NCHUNK (DIM / KCHUNK)
#define LDS_STRIDE 36   // (36*r + c) % 64 distinct for 16 consecutive rows; 36*4B = 144B (16B aligned)
#define SIM_STRIDE 132  // (132*r + j) % 64 == (4r + j) % 64 -> conflict-free row scan
#define QCAP 32         // per-row candidate queue capacity (overflow -> direct rescan)

// ---------------------------------------------------------------------------
// Phase 1: cosine-normalize rows.  One wave (32 lanes) per row.
// ---------------------------------------------------------------------------
__global__ __launch_bounds__(256) void normalize_rows(const float* __restrict__ f,
                                                      float* __restrict__ fn)
{
    const int wave = threadIdx.x >> 5;
    const int lane = threadIdx.x & 31;
    const int row  = blockIdx.x * 8 + wave;
    const float* src = f  + (size_t)row * DIM;
    float*       dst = fn + (size_t)row * DIM;

    float s = 0.0f;
    for (int j = lane; j < DIM; j += 32) { float v = src[j]; s += v * v; }
    for (int off = 16; off > 0; off >>= 1) s += __shfl_xor(s, off, 32);
    const float inv = 1.0f / (sqrtf(s) + 1e-10f);
    for (int j = lane; j < DIM; j += 32) dst[j] = src[j] * inv;
}

// ---------------------------------------------------------------------------
// Async global->LDS copy (CDNA5 TDM-class path, tracked by ASYNCcnt).
// LDS destination address = low 32 bits of the generic pointer.
// ---------------------------------------------------------------------------
__device__ __forceinline__ void async_b128(const float* lds_p, const float* g)
{
    asm volatile("global_load_async_to_lds_b128 %0, %1, off"
                 :: "v"((unsigned)(uintptr_t)lds_p),
                    "v"((unsigned long long)(uintptr_t)g)
                 : "memory");
}

// One staging group: B panel chunk (128 cols x 32 k) + A chunk (16 rows x 32 k).
// Every wave issues exactly 5 async instructions (4 B all-lanes + 1 A even-lanes),
// so ASYNCcnt bookkeeping is wave-uniform.
__device__ __forceinline__ void issue_group(const float* __restrict__ fn,
                                            int rbase, int colbase, int kb,
                                            float* __restrict__ Ab,
                                            float* __restrict__ Bb, int t)
{
    #pragma unroll
    for (int i = 0; i < 4; ++i) {
        const int idx = t + i * 256;            // 0..1023 float4 slots
        const int col = idx >> 3, q = idx & 7;
        async_b128(&Bb[col * LDS_STRIDE + q * 4],
                   fn + (size_t)(colbase + col) * DIM + kb + q * 4);
    }
    {
        const int half = t >> 1;                // 0..127 float4 slots
        const int row = half >> 3, q = half & 7;
        if ((t & 1) == 0)                       // 16 active lanes in EVERY wave
            async_b128(&Ab[row * LDS_STRIDE + q * 4],
                       fn + (size_t)(rbase + row) * DIM + kb + q * 4);
    }
}

// ---------------------------------------------------------------------------
// Phase 2: fused sim = fn @ fn^T (f32 WMMA, dual accumulator chains) with
// double-buffered async LDS staging pipelined across chunks AND panels,
// plus running top-16 per row.
// ---------------------------------------------------------------------------
__global__ __launch_bounds__(256) void simtopk(const float* __restrict__ fn,
                                               long long* __restrict__ out)
{
    __shared__ float Alds[2][ROWS_PER_BLOCK * LDS_STRIDE];
    __shared__ float Blds[2][PANEL * LDS_STRIDE];
    __shared__ float simbuf[ROWS_PER_BLOCK * SIM_STRIDE];
    __shared__ float qval[ROWS_PER_BLOCK][QCAP];
    __shared__ int   qidx[ROWS_PER_BLOCK][QCAP];
    __shared__ int   qcount[ROWS_PER_BLOCK];
    __shared__ float topval[ROWS_PER_BLOCK][TOPK + 1];
    __shared__ int   topidx[ROWS_PER_BLOCK][TOPK + 1];
    __shared__ float rowmin[ROWS_PER_BLOCK];

    const int t     = threadIdx.x;
    const int wave  = t >> 5;
    const int lane  = t & 31;
    const int rbase = blockIdx.x * ROWS_PER_BLOCK;

    if (t < ROWS_PER_BLOCK) {
        rowmin[t] = -FLT_MAX;
        for (int i = 0; i < TOPK; ++i) { topval[t][i] = -FLT_MAX; topidx[t][i] = 0; }
    }

    int buf = 0;
    issue_group(fn, rbase, /*colbase=*/0, /*kb=*/0, Alds[0], Blds[0], t); // prologue

    for (int p = 0; p < N_NODES / PANEL; ++p) {
        const int colbase = p * PANEL;
        v8f acc0 = {}, acc1 = {};

        for (int c = 0; c < NCHUNK; ++c) {
            const int  kb          = c * KCHUNK;
            const bool lastc       = (c == NCHUNK - 1);
            const bool last_global = lastc && (p == N_NODES / PANEL - 1);

            // Prefetch next chunk (or next panel's first chunk) into the other buffer.
            if (!last_global) {
                const int ncol = lastc ? colbase + PANEL : colbase;
                const int nkb  = lastc ? 0 : kb + KCHUNK;
                issue_group(fn, rbase, ncol, nkb, Alds[buf ^ 1], Blds[buf ^ 1], t);
            }
            // Own group for `buf` is complete once <=5 (the just-issued group) remain.
            if (last_global) asm volatile("s_wait_asynccnt 0x0" ::: "memory");
            else             asm volatile("s_wait_asynccnt 0x5" ::: "memory");
            __syncthreads();        // all waves' async data landed in LDS

            const float* Ab = Alds[buf];
            const float* Bb = Blds[buf];
            // ISA 32-bit A 16x4 / B 4x16 layout: lanes 0-15 -> K+{0,1}, lanes 16-31 -> K+{2,3}
            const int m    = lane & 15;
            const int ksel = (lane < 16) ? 0 : 2;
            const int cl   = wave * 16 + (lane & 15);
            #pragma unroll
            for (int kk = 0; kk < KCHUNK / 4; ++kk) {
                const int k4 = kk * 4;
                v2f a = *(const v2f*)&Ab[m  * LDS_STRIDE + k4 + ksel];
                v2f b = *(const v2f*)&Bb[cl * LDS_STRIDE + k4 + ksel];
                if (kk & 1)
                    acc1 = __builtin_amdgcn_wmma_f32_16x16x4_f32(false, a, false, b,
                                                                 (short)0, acc1, false, false);
                else
                    acc0 = __builtin_amdgcn_wmma_f32_16x16x4_f32(false, a, false, b,
                                                                 (short)0, acc0, false, false);
            }
            __syncthreads();        // everyone done reading `buf` before it is refilled
            buf ^= 1;
        }

        const v8f acc = acc0 + acc1;

        // Spill the 16x16 f32 tile (C layout: vgpr v = row v (lanes<16) / v+8 (lanes>=16))
        {
            const int rhalf = (lane < 16) ? 0 : 8;
            const int cl    = wave * 16 + (lane & 15);
            #pragma unroll
            for (int v = 0; v < 8; ++v)
                simbuf[(v + rhalf) * SIM_STRIDE + cl] = acc[v];
        }
        if (t < ROWS_PER_BLOCK) qcount[t] = 0;
        __syncthreads();

        // Parallel threshold filter: 16 threads/row, 8 candidates each.
        // rowmin is last panel's min -> conservative (only over-admits).
        {
            const int   r    = t & 15;
            const int   jb   = (t >> 4) * 8;
            const int   grow = rbase + r;
            const float rmin = rowmin[r];
            for (int j = jb; j < jb + 8; ++j) {
                float v = simbuf[r * SIM_STRIDE + j];
                const int gcol = colbase + j;
                if (gcol == grow) v = 0.0f;          // reference zeroes the diagonal
                if (v > rmin) {
                    const int pos = atomicAdd(&qcount[r], 1);
                    if (pos < QCAP) { qval[r][pos] = v; qidx[r][pos] = gcol; }
                }
            }
        }
        __syncthreads();

        // Serial per-row insert; queue overflow -> rescan the LDS sim row directly.
        if (t < ROWS_PER_BLOCK) {
            const int r    = t;
            const int grow = rbase + r;
            float mval = topval[r][0]; int mslot = 0;
            for (int i = 1; i < TOPK; ++i)
                if (topval[r][i] < mval) { mval = topval[r][i]; mslot = i; }
            const int cnt = qcount[r];
            if (cnt > QCAP) {
                for (int j = 0; j < PANEL; ++j) {
                    float v = simbuf[r * SIM_STRIDE + j];
                    const int gcol = colbase + j;
                    if (gcol == grow) v = 0.0f;
                    if (v > mval) {
                        topval[r][mslot] = v; topidx[r][mslot] = gcol;
                        mval = topval[r][0]; mslot = 0;
                        for (int i = 1; i < TOPK; ++i)
                            if (topval[r][i] < mval) { mval = topval[r][i]; mslot = i; }
                    }
                }
            } else {
                for (int e = 0; e < cnt; ++e) {
                    const float v = qval[r][e];
                    if (v > mval) {
                        topval[r][mslot] = v; topidx[r][mslot] = qidx[r][e];
                        mval = topval[r][0]; mslot = 0;
                        for (int i = 1; i < TOPK; ++i)
                            if (topval[r][i] < mval) { mval = topval[r][i]; mslot = i; }
                    }
                }
            }
            rowmin[r] = mval;
        }
        // next loop iteration's first __syncthreads() separates these phases
        // from LDS buffer refills; async prefetch for panel p+1 is already in flight.
    }
    __syncthreads();

    // Sort descending (ties -> smaller index, matching lax.top_k) and emit int64 edges.
    if (t < ROWS_PER_BLOCK) {
        const int r    = t;
        const int grow = rbase + r;
        for (int i = 0; i < TOPK; ++i) {
            int best = i;
            for (int j = i + 1; j < TOPK; ++j) {
                if (topval[r][j] > topval[r][best] ||
                    (topval[r][j] == topval[r][best] && topidx[r][j] < topidx[r][best]))
                    best = j;
            }
            const float tv = topval[r][i]; topval[r][i] = topval[r][best]; topval[r][best] = tv;
            const int   ti = topidx[r][i]; topidx[r][i] = topidx[r][best]; topidx[r][best] = ti;
        }
        long long* orow = out + (size_t)grow * TOPK;
        long long* ocol = out + (size_t)N_NODES * TOPK + (size_t)grow * TOPK;
        for (int i = 0; i < TOPK; ++i) { orow[i] = grow; ocol[i] = topidx[r][i]; }
    }
}

// ---------------------------------------------------------------------------
extern "C" void kernel_launch(void* const* d_in, const int* in_sizes, int n_in,
                              void* d_out, int out_size, void* d_ws, size_t ws_size,
                              hipStream_t stream) {
    const float* feature = (const float*)d_in[0];   // [16384, 512] f32
    // d_in[1] is k (==16), baked in as TOPK.
    float* fn = (float*)d_ws;                       // 32 MB normalized features

    normalize_rows<<<N_NODES / 8, 256, 0, stream>>>(feature, fn);
    simtopk<<<N_NODES / ROWS_PER_BLOCK, 256, 0, stream>>>(fn, (long long*)d_out);
}